// GINConv_4363686772848
// MI455X (gfx1250) — compile-verified
//
#include <hip/hip_runtime.h>

// GIN layer for MI455X (gfx1250, wave32):
//   agg = x + segment_sum(x[src], dst)          (edge scatter, f32 atomics, L2-resident)
//   out = relu(agg @ W1 + b1) @ W2 + b2         (fused 2-GEMM MLP via V_WMMA_F32_16X16X4_F32)

typedef __attribute__((ext_vector_type(2))) float v2f;
typedef __attribute__((ext_vector_type(8))) float v8f;

#define N_NODES 50000
#define N_EDGES 640000
#define DIM 128
#define LDS_STRIDE 132          // 128 + 4 pad: spreads row accesses across LDS banks
#define M_TILES (N_NODES / 16)  // 3125, exact
#define TILES_PER_WG 5
#define N_WG (M_TILES / TILES_PER_WG)  // 625

// ---------------------------------------------------------------- agg = x
__global__ __launch_bounds__(256) void gin_init_agg(const float4* __restrict__ x,
                                                    float4* __restrict__ agg) {
    int tid = blockIdx.x * blockDim.x + threadIdx.x;
    if (tid < N_NODES * DIM / 4) agg[tid] = x[tid];
}

// ------------------------------------------------- agg[dst] += x[src] per edge
// One wave32 per edge; lane L owns floats [4L, 4L+4). Coalesced 512B gather,
// native global_atomic_add_f32 (relaxed, device scope) into L2-resident agg.
__global__ __launch_bounds__(256) void gin_scatter(const float* __restrict__ x,
                                                   const int* __restrict__ src,
                                                   const int* __restrict__ dst,
                                                   float* __restrict__ agg) {
    int tid = blockIdx.x * blockDim.x + threadIdx.x;
    int e = tid >> 5;
    int lane = tid & 31;
    if (e >= N_EDGES) return;
    int s = src[e];
    int d = dst[e];
    const float4 v = ((const float4*)(x + (size_t)s * DIM))[lane];
    float* p = agg + (size_t)d * DIM + lane * 4;
    __hip_atomic_fetch_add(p + 0, v.x, __ATOMIC_RELAXED, __HIP_MEMORY_SCOPE_AGENT);
    __hip_atomic_fetch_add(p + 1, v.y, __ATOMIC_RELAXED, __HIP_MEMORY_SCOPE_AGENT);
    __hip_atomic_fetch_add(p + 2, v.z, __ATOMIC_RELAXED, __HIP_MEMORY_SCOPE_AGENT);
    __hip_atomic_fetch_add(p + 3, v.w, __ATOMIC_RELAXED, __HIP_MEMORY_SCOPE_AGENT);
}

// ------------------------------------- out = relu(agg @ W1 + b1) @ W2 + b2
// 8 waves per block; wave w owns output columns [16w, 16w+16).
// Each wave preloads its W1/W2 column-slice B-fragments into registers once
// (32 v2f each), then loops over TILES_PER_WG row tiles of 16 nodes.
// A-fragments are prefetched into registers as one DS clause so the 32 WMMAs
// of each GEMM issue back-to-back (accumulator chaining has no hazard).
__global__ __launch_bounds__(256) void gin_mlp(const float* __restrict__ agg,
                                               const float* __restrict__ W1,
                                               const float* __restrict__ b1,
                                               const float* __restrict__ W2,
                                               const float* __restrict__ b2,
                                               float* __restrict__ out) {
    __shared__ float lds_a[16 * LDS_STRIDE];
    __shared__ float lds_h[16 * LDS_STRIDE];

    const int tid  = threadIdx.x;
    const int lane = tid & 31;
    const int wid  = tid >> 5;       // wave 0..7 -> N slice
    const int nl   = lane & 15;      // N (for B/C) or M (for A) within tile
    const int half = lane >> 4;      // 0: K=k,k+1  1: K=k+2,k+3
    const int kp   = half * 2;
    const int n    = wid * 16 + nl;  // global output column 0..127

    // --- B fragments: lane holds column n, K-pair (4k+kp, 4k+kp+1) in v2f[k]
    v2f w1f[32], w2f[32];
#pragma unroll
    for (int k = 0; k < 32; ++k) {
        int kk = 4 * k + kp;
        w1f[k].x = W1[(size_t)kk * DIM + n];
        w1f[k].y = W1[(size_t)(kk + 1) * DIM + n];
        w2f[k].x = W2[(size_t)kk * DIM + n];
        w2f[k].y = W2[(size_t)(kk + 1) * DIM + n];
    }
    const float bias1 = b1[n];
    const float bias2 = b2[n];

    for (int t = 0; t < TILES_PER_WG; ++t) {
        const int tile = blockIdx.x * TILES_PER_WG + t;
        const int row_base = tile * 16;

        // cooperative load: 16x128 agg tile -> padded LDS (512 float4, 2/thread)
        const float4* g = (const float4*)(agg + (size_t)row_base * DIM);
#pragma unroll
        for (int i = 0; i < 2; ++i) {
            int c  = tid + i * 256;  // 0..511
            int r  = c >> 5;         // row 0..15
            int c4 = c & 31;         // float4 column
            float4 v = g[r * 32 + c4];
            float* dp = &lds_a[r * LDS_STRIDE + c4 * 4];
            dp[0] = v.x; dp[1] = v.y; dp[2] = v.z; dp[3] = v.w;
        }
        __syncthreads();  // B1: lds_a ready; also fences prev iter's lds_h reads

        // GEMM1: prefetch all A fragments, then 32 chained WMMAs
        {
            v2f af[32];
#pragma unroll
            for (int k = 0; k < 32; ++k) {
                const float* ap = &lds_a[nl * LDS_STRIDE + 4 * k + kp];
                af[k].x = ap[0]; af[k].y = ap[1];
            }
            v8f c1 = {};
#pragma unroll
            for (int k = 0; k < 32; ++k)
                c1 = __builtin_amdgcn_wmma_f32_16x16x4_f32(
                    false, af[k], false, w1f[k], (short)0, c1, false, false);

            // bias + relu, scatter C layout (M = r + 8*half, N = n) into lds_h
#pragma unroll
            for (int r = 0; r < 8; ++r) {
                float h = c1[r] + bias1;
                h = h > 0.0f ? h : 0.0f;
                lds_h[(r + half * 8) * LDS_STRIDE + n] = h;
            }
        }
        __syncthreads();  // B2: lds_h ready

        // GEMM2: same pattern from lds_h
        {
            v2f af[32];
#pragma unroll
            for (int k = 0; k < 32; ++k) {
                const float* ap = &lds_h[nl * LDS_STRIDE + 4 * k + kp];
                af[k].x = ap[0]; af[k].y = ap[1];
            }
            v8f c2 = {};
#pragma unroll
            for (int k = 0; k < 32; ++k)
                c2 = __builtin_amdgcn_wmma_f32_16x16x4_f32(
                    false, af[k], false, w2f[k], (short)0, c2, false, false);

            // out is write-once: non-temporal stores keep agg/x/W hot in L2
#pragma unroll
            for (int r = 0; r < 8; ++r) {
                __builtin_nontemporal_store(
                    c2[r] + bias2,
                    &out[(size_t)(row_base + r + half * 8) * DIM + n]);
            }
        }
    }
}

extern "C" void kernel_launch(void* const* d_in, const int* in_sizes, int n_in,
                              void* d_out, int out_size, void* d_ws, size_t ws_size,
                              hipStream_t stream) {
    const float* x  = (const float*)d_in[0];
    const int*   ei = (const int*)d_in[1];   // [2, N_EDGES]: src row then dst row
    const float* W1 = (const float*)d_in[2];
    const float* b1 = (const float*)d_in[3];
    const float* W2 = (const float*)d_in[4];
    const float* b2 = (const float*)d_in[5];
    float* out = (float*)d_out;
    float* agg = (float*)d_ws;               // 50000*128*4 = 25.6 MB scratch

    gin_init_agg<<<(N_NODES * DIM / 4) / 256, 256, 0, stream>>>(
        (const float4*)x, (float4*)agg);

    gin_scatter<<<(N_EDGES * 32) / 256, 256, 0, stream>>>(
        x, ei, ei + N_EDGES, agg);

    gin_mlp<<<N_WG, 256, 0, stream>>>(agg, W1, b1, W2, b2, out);
}